// KLDivLossColBERT_74466142978304
// MI455X (gfx1250) — compile-verified
//
#include <hip/hip_runtime.h>
#include <hip/hip_bf16.h>

typedef float v2f __attribute__((ext_vector_type(2)));
typedef float v8f __attribute__((ext_vector_type(8)));
typedef int   v4i __attribute__((ext_vector_type(4)));

#define NEGVAL  (-9999.0f)
#define LQ      64
#define LD      256
#define DDIM    128
#define NWAY    16
#define CHUNK   64              // D rows per LDS chunk
#define NCHUNK  (LD / CHUNK)
#define STRIDE  132             // 132 % 64 == 4 -> conflict-free columns, 16B aligned rows

#if __has_builtin(__builtin_amdgcn_global_load_async_to_lds_b128)
#define ASYNC_LDS 1
typedef __attribute__((address_space(1))) v4i* gasptr;   // global int4*
typedef __attribute__((address_space(3))) v4i* lasptr;   // LDS int4*
#endif

// Stage one 64x128 fp32 chunk (global, row-major) into padded LDS.
// Async path: global_load_async_to_lds_b128, per-lane LDS address keeps the
// padded stride; tracked by ASYNCcnt so WMMA on the other buffer overlaps.
static __device__ __forceinline__
void chunk_stage(const float* __restrict__ src, float* __restrict__ dstLds, int tid)
{
#ifdef ASYNC_LDS
    for (int i = tid; i < (CHUNK * DDIM) / 4; i += 256) {
        int e = i * 4;
        int row = e >> 7;              // / 128
        int col = e & (DDIM - 1);
        __builtin_amdgcn_global_load_async_to_lds_b128(
            (gasptr)(src + e), (lasptr)&dstLds[row * STRIDE + col], 0, 0);
    }
#else
    for (int i = tid; i < (CHUNK * DDIM) / 4; i += 256) {
        float4 v = ((const float4*)src)[i];
        int e = i * 4;
        int row = e >> 7;
        int col = e & (DDIM - 1);
        *(float4*)&dstLds[row * STRIDE + col] = v;
    }
#endif
}

static __device__ __forceinline__ void chunk_wait()
{
#ifdef ASYNC_LDS
#if __has_builtin(__builtin_amdgcn_s_wait_asynccnt)
    __builtin_amdgcn_s_wait_asynccnt(0);
#else
    asm volatile("s_wait_asynccnt 0x0" ::: "memory");
#endif
#endif
}

// ---------------------------------------------------------------------------
// Kernel 1: per-(b,n) MaxSim score via fp32 WMMA GEMM (64x256x128),
// double-buffered async D-chunk staging.
// ---------------------------------------------------------------------------
__global__ __launch_bounds__(256)
void colbert_maxsim_kernel(const float* __restrict__ q_reps,
                           const float* __restrict__ d_reps,
                           const int*   __restrict__ d_mask,
                           float*       __restrict__ scores)
{
    extern __shared__ float smem[];
    float* Qs      = smem;                       // 64*132
    float* Ds0     = Qs  + LQ * STRIDE;          // 64*132 (buffer 0)
    float* Ds1     = Ds0 + CHUNK * STRIDE;       // 64*132 (buffer 1)
    float* qinv    = Ds1 + CHUNK * STRIDE;       // 64
    float* dinv    = qinv + LQ;                  // 64
    float* partmax = dinv + CHUNK;               // 64*4
    int*   smask   = (int*)(partmax + LQ * 4);   // 64
    float* rowred  = (float*)(smask + CHUNK);    // 64

    const int tid  = threadIdx.x;
    const int wave = tid >> 5;
    const int lane = tid & 31;
    const int half = lane >> 4;        // 0: lanes 0-15, 1: lanes 16-31
    const int l16  = lane & 15;

    const int b = blockIdx.x / NWAY;
    const int n = blockIdx.x % NWAY;

    const float* dbase = d_reps + (size_t)(b * NWAY + n) * LD * DDIM;
    const int*   mbase = d_mask + (size_t)(b * NWAY + n) * LD;

    // ---- stage Q [64x128] into LDS; kick off async load of D chunk 0 ----
    const float* qsrc = q_reps + (size_t)b * LQ * DDIM;
    for (int i = tid; i < (LQ * DDIM) / 4; i += 256) {
        float4 v = ((const float4*)qsrc)[i];
        int e = i * 4;
        int row = e >> 7;
        int col = e & (DDIM - 1);
        *(float4*)&Qs[row * STRIDE + col] = v;
    }
    chunk_stage(dbase, Ds0, tid);                // chunk 0 -> buffer 0 (async)
    for (int i = tid; i < LQ * 4; i += 256) partmax[i] = NEGVAL;
    __syncthreads();

    // ---- query row inverse norms ----
    if (tid < LQ) {
        float ss = 0.0f;
        for (int k = 0; k < DDIM; ++k) { float x = Qs[tid * STRIDE + k]; ss += x * x; }
        qinv[tid] = 1.0f / fmaxf(sqrtf(ss), 1e-12f);
    }

    // wave w owns 16x16 tiles w and w+8 of the 4x4 tile grid
    const int mt0 = wave >> 2,       nt0 = wave & 3;
    const int mt1 = (wave + 8) >> 2, nt1 = (wave + 8) & 3;

    for (int c = 0; c < NCHUNK; ++c) {
        chunk_wait();        // this wave's async loads for chunk c complete
        __syncthreads();     // whole chunk c visible; prior compute done

        float* Dsc = (c & 1) ? Ds1 : Ds0;
        if (c + 1 < NCHUNK)  // overlap: stream chunk c+1 while computing on c
            chunk_stage(dbase + (size_t)(c + 1) * CHUNK * DDIM,
                        (c & 1) ? Ds0 : Ds1, tid);

        if (tid < CHUNK) {
            smask[tid] = mbase[c * CHUNK + tid];
            float ss = 0.0f;
            for (int k = 0; k < DDIM; ++k) { float x = Dsc[tid * STRIDE + k]; ss += x * x; }
            dinv[tid] = 1.0f / fmaxf(sqrtf(ss), 1e-12f);
        }
        __syncthreads();

        // ---- fp32 WMMA GEMM: two 16x16 tiles per wave, K = 128 in steps of 4 ----
        v8f acc0 = {0.f,0.f,0.f,0.f,0.f,0.f,0.f,0.f};
        v8f acc1 = {0.f,0.f,0.f,0.f,0.f,0.f,0.f,0.f};
        const int ar0 = (mt0 * 16 + l16) * STRIDE;
        const int ar1 = (mt1 * 16 + l16) * STRIDE;
        const int br0 = (nt0 * 16 + l16) * STRIDE;
        const int br1 = (nt1 * 16 + l16) * STRIDE;
        for (int k = 0; k < DDIM; k += 4) {
            const int kk = k + 2 * half;           // lanes 0-15: K=k,k+1; 16-31: K=k+2,k+3
            v2f a0 = *(const v2f*)&Qs[ar0 + kk];
            v2f b0 = *(const v2f*)&Dsc[br0 + kk];
            v2f a1 = *(const v2f*)&Qs[ar1 + kk];
            v2f b1 = *(const v2f*)&Dsc[br1 + kk];
            acc0 = __builtin_amdgcn_wmma_f32_16x16x4_f32(false, a0, false, b0,
                                                         (short)0, acc0, false, false);
            acc1 = __builtin_amdgcn_wmma_f32_16x16x4_f32(false, a1, false, b1,
                                                         (short)0, acc1, false, false);
        }

        // ---- epilogue: scale by inv-norms, mask, per-row max ----
        {
            const int   nn = nt0 * 16 + l16;
            const float di = dinv[nn];
            const int   mk = smask[nn];
            #pragma unroll
            for (int r = 0; r < 8; ++r) {
                int   m = mt0 * 16 + r + 8 * half;
                float v = acc0[r] * qinv[m] * di;
                if (mk == 0) v = NEGVAL;
                #pragma unroll
                for (int off = 1; off < 16; off <<= 1) v = fmaxf(v, __shfl_xor(v, off, 32));
                if (l16 == 0) partmax[m * 4 + nt0] = fmaxf(partmax[m * 4 + nt0], v);
            }
        }
        {
            const int   nn = nt1 * 16 + l16;
            const float di = dinv[nn];
            const int   mk = smask[nn];
            #pragma unroll
            for (int r = 0; r < 8; ++r) {
                int   m = mt1 * 16 + r + 8 * half;
                float v = acc1[r] * qinv[m] * di;
                if (mk == 0) v = NEGVAL;
                #pragma unroll
                for (int off = 1; off < 16; off <<= 1) v = fmaxf(v, __shfl_xor(v, off, 32));
                if (l16 == 0) partmax[m * 4 + nt1] = fmaxf(partmax[m * 4 + nt1], v);
            }
        }
    }
    __syncthreads();

    // ---- score = sum over 64 query rows of (max over Ld) ----
    if (tid < LQ) {
        float mx = fmaxf(fmaxf(partmax[tid * 4 + 0], partmax[tid * 4 + 1]),
                         fmaxf(partmax[tid * 4 + 2], partmax[tid * 4 + 3]));
        rowred[tid] = mx;
    }
    __syncthreads();
    if (tid < 32) {
        float s = rowred[tid] + rowred[tid + 32];
        #pragma unroll
        for (int off = 16; off > 0; off >>= 1) s += __shfl_xor(s, off, 32);
        if (tid == 0) scores[blockIdx.x] = s;
    }
}

// ---------------------------------------------------------------------------
// Kernel 2: softmax-log over NWAY + KLDiv(batchmean, log_target=True)
// ---------------------------------------------------------------------------
__global__ __launch_bounds__(32)
void colbert_kl_kernel(const float* __restrict__ scores,
                       const float* __restrict__ labels,
                       float*       __restrict__ out,
                       int B)
{
    const int bb = threadIdx.x;          // one lane per batch row (B == 32)
    float kl = 0.0f;
    if (bb < B) {
        const float* s   = scores + bb * NWAY;
        const float* lab = labels + bb * NWAY;
        float m = -3.402823466e38f;
        for (int k = 0; k < NWAY; ++k) m = fmaxf(m, s[k]);
        float se = 0.0f;
        for (int k = 0; k < NWAY; ++k) se += expf(s[k] - m);
        float lse = logf(se) + m;
        for (int k = 0; k < NWAY; ++k) {
            float lp = s[k] - lse;       // log_softmax(scores)
            float l  = lab[k];
            kl += expf(l) * (l - lp);
        }
    }
    #pragma unroll
    for (int off = 16; off > 0; off >>= 1) kl += __shfl_xor(kl, off, 32);
    if (bb == 0) out[0] = kl / (float)B;
}

// ---------------------------------------------------------------------------
extern "C" void kernel_launch(void* const* d_in, const int* in_sizes, int n_in,
                              void* d_out, int out_size, void* d_ws, size_t ws_size,
                              hipStream_t stream)
{
    const float* q_reps = (const float*)d_in[0];   // [B, 64, 128] f32
    const float* d_reps = (const float*)d_in[1];   // [B, 16, 256, 128] f32
    const int*   d_mask = (const int*)d_in[2];     // [B, 16, 256] i32
    const float* labels = (const float*)d_in[3];   // [B, 16] f32
    float* out = (float*)d_out;
    float* scores_ws = (float*)d_ws;               // [B*NWAY] floats

    const int nBN = in_sizes[3];                   // B * NWAY (= 512)
    const int B   = nBN / NWAY;

    const size_t lds_bytes =
        (size_t)(LQ * STRIDE + 2 * CHUNK * STRIDE + LQ + CHUNK + LQ * 4 + CHUNK + LQ)
        * sizeof(float);

    colbert_maxsim_kernel<<<dim3(nBN), dim3(256), lds_bytes, stream>>>(
        q_reps, d_reps, d_mask, scores_ws);
    colbert_kl_kernel<<<dim3(1), dim3(32), 0, stream>>>(
        scores_ws, labels, out, B);
}